// GAT_4260607557858
// MI455X (gfx1250) — compile-verified
//
#include <hip/hip_runtime.h>
#include <hip/hip_bf16.h>

// ---------------------------------------------------------------------------
// GATv2 + GatedGraphConv stack on MI455X (gfx1250).
// Dense GEMMs use V_WMMA_F32_16X16X4_F32 (native fp32 matrix op, wave32).
// Edge phases are fused wave-per-edge kernels (memory-bound part).
// ---------------------------------------------------------------------------

typedef __attribute__((ext_vector_type(2))) float v2f;
typedef __attribute__((ext_vector_type(8))) float v8f;

#define NODES 50000
#define EDGES 800000
#define DIM   64
#define NEG_SLOPE 0.2f

// ---- ordered-uint encoding so atomicMax works on signed floats -------------
__device__ __forceinline__ unsigned encF(float f) {
    unsigned u = __float_as_uint(f);
    return (u & 0x80000000u) ? ~u : (u | 0x80000000u);
}
__device__ __forceinline__ float decF(unsigned u) {
    unsigned v = (u & 0x80000000u) ? (u & 0x7FFFFFFFu) : ~u;
    return __uint_as_float(v);
}

// ---------------------------------------------------------------------------
// Generic C[M x Nout] = A[M x 64] * W^T + bias, W stored (Nout x 64) row-major.
// One wave32 per 16x16 output tile; 16 k-steps of v_wmma_f32_16x16x4_f32.
// M must be a multiple of 16 (50000 = 3125*16), Nout a multiple of 16.
// ---------------------------------------------------------------------------
__global__ void __launch_bounds__(256)
gemm_bias_wmma(const float* __restrict__ A, const float* __restrict__ W,
               const float* __restrict__ bias, float* __restrict__ C,
               int M, int K, int Nout)
{
    const int lane = threadIdx.x & 31;
    const int wave = blockIdx.x * 8 + (threadIdx.x >> 5);
    const int colTiles = Nout >> 4;
    const int total = (M >> 4) * colTiles;
    if (wave >= total) return;                 // wave-uniform: EXEC all-ones below

    const int rt   = wave / colTiles;
    const int ct   = wave - rt * colTiles;
    const int row0 = rt << 4;
    const int col0 = ct << 4;
    const int lr   = lane & 15;
    const int hi   = lane >> 4;                // 0: K pair {0,1}, 1: {2,3}

    const float* __restrict__ arow = A + (size_t)(row0 + lr) * K;
    const float* __restrict__ wrow = W + (size_t)(col0 + lr) * K;

    v8f acc = {};
    #pragma unroll
    for (int k0 = 0; k0 < 64; k0 += 4) {
        v2f a, b;
        a.x = arow[k0 + 2 * hi];
        a.y = arow[k0 + 2 * hi + 1];
        b.x = wrow[k0 + 2 * hi];               // B[k][n] = W[n][k]
        b.y = wrow[k0 + 2 * hi + 1];
        acc = __builtin_amdgcn_wmma_f32_16x16x4_f32(
                  false, a, false, b, (short)0, acc, false, false);
    }

    const float bv = bias[col0 + lr];
    float* crow = C + (size_t)(row0 + hi * 8) * Nout + col0 + lr;
    #pragma unroll
    for (int v = 0; v < 8; ++v)
        crow[(size_t)v * Nout] = acc[v] + bv;
}

// ---------------------------------------------------------------------------
// Fused GATv2 edge pass 1: score_e = sum_k leaky_relu(el[s]+er[d])*attn[k],
// plus running segment max via ordered-uint atomicMax. One wave per edge.
// ---------------------------------------------------------------------------
__global__ void __launch_bounds__(256)
edge_score(const float* __restrict__ el, const float* __restrict__ er,
           const float* __restrict__ attn, const int* __restrict__ src,
           const int* __restrict__ dst, float* __restrict__ score,
           unsigned* __restrict__ mEnc)
{
    const int  lane = threadIdx.x & 31;
    const long e    = (long)blockIdx.x * 8 + (threadIdx.x >> 5);
    const long Ef   = (long)EDGES + NODES;
    if (e >= Ef) return;
    int s, d;
    if (e < EDGES) { s = src[e]; d = dst[e]; } else { s = d = (int)(e - EDGES); }

    float p = 0.f;
    #pragma unroll
    for (int j = 0; j < 2; ++j) {
        const int k = lane + 32 * j;
        float v = el[(size_t)s * DIM + k] + er[(size_t)d * DIM + k];
        v = v > 0.f ? v : NEG_SLOPE * v;
        p += v * attn[k];
    }
    #pragma unroll
    for (int off = 16; off; off >>= 1) p += __shfl_xor(p, off, 32);
    if (lane == 0) {
        score[e] = p;
        atomicMax(&mEnc[d], encF(p));
    }
}

// ---- pass 2: ex = exp(score - max[dst]); den[dst] += ex (thread/edge) ------
__global__ void edge_exp(float* __restrict__ score,
                         const unsigned* __restrict__ mEnc,
                         float* __restrict__ den, const int* __restrict__ dst)
{
    const long e  = (long)blockIdx.x * blockDim.x + threadIdx.x;
    const long Ef = (long)EDGES + NODES;
    if (e >= Ef) return;
    const int d  = (e < EDGES) ? dst[e] : (int)(e - EDGES);
    const float ex = expf(score[e] - decF(mEnc[d]));
    score[e] = ex;                              // reuse buffer: score -> ex
    atomicAdd(&den[d], ex);
}

// ---- pass 3: acc[d] += (ex/den[d]) * el[s]  (wave per edge) ----------------
__global__ void __launch_bounds__(256)
edge_aggregate(const float* __restrict__ el, const float* __restrict__ ex,
               const float* __restrict__ den, const int* __restrict__ src,
               const int* __restrict__ dst, float* __restrict__ acc)
{
    const int  lane = threadIdx.x & 31;
    const long e    = (long)blockIdx.x * 8 + (threadIdx.x >> 5);
    const long Ef   = (long)EDGES + NODES;
    if (e >= Ef) return;
    int s, d;
    if (e < EDGES) { s = src[e]; d = dst[e]; } else { s = d = (int)(e - EDGES); }
    const float a = ex[e] / den[d];
    #pragma unroll
    for (int j = 0; j < 2; ++j) {
        const int k = lane + 32 * j;
        atomicAdd(&acc[(size_t)d * DIM + k], a * el[(size_t)s * DIM + k]);
    }
}

// ---- elu, write both persistent h and the embedding slot -------------------
__global__ void elu_write(const float* __restrict__ acc, float* __restrict__ h,
                          float* __restrict__ emb, long n)
{
    const long i = (long)blockIdx.x * blockDim.x + threadIdx.x;
    if (i >= n) return;
    const float v = acc[i];
    const float o = v > 0.f ? v : (expf(v) - 1.f);
    h[i] = o;
    emb[i] = o;
}

// ---- GGC message scatter: acc[d] += trans[etype][s]  (wave per edge) -------
__global__ void __launch_bounds__(256)
ggc_scatter(const float* __restrict__ t0, const float* __restrict__ t1,
            const int* __restrict__ ety, const int* __restrict__ src,
            const int* __restrict__ dst, float* __restrict__ acc)
{
    const int  lane = threadIdx.x & 31;
    const long e    = (long)blockIdx.x * 8 + (threadIdx.x >> 5);
    const long Ef   = (long)EDGES + NODES;
    if (e >= Ef) return;
    int s, d;
    if (e < EDGES) { s = src[e]; d = dst[e]; } else { s = d = (int)(e - EDGES); }
    const float* __restrict__ tr = (ety[e] == 0) ? t0 : t1;
    #pragma unroll
    for (int j = 0; j < 2; ++j) {
        const int k = lane + 32 * j;
        atomicAdd(&acc[(size_t)d * DIM + k], tr[(size_t)s * DIM + k]);
    }
}

// ---- GRU cell update (thread per (n,d) element) ----------------------------
__global__ void gru_update(const float* __restrict__ gi,
                           const float* __restrict__ gh,
                           float* __restrict__ h, float* __restrict__ emb,
                           long nElem)
{
    const long i = (long)blockIdx.x * blockDim.x + threadIdx.x;
    if (i >= nElem) return;
    const long n = i >> 6;
    const int  d = (int)(i & 63);
    const float* gin = gi + n * (3 * DIM);
    const float* ghn = gh + n * (3 * DIM);
    const float r = 1.f / (1.f + expf(-(gin[d]           + ghn[d])));
    const float z = 1.f / (1.f + expf(-(gin[DIM + d]     + ghn[DIM + d])));
    const float c = tanhf(gin[2 * DIM + d] + r * ghn[2 * DIM + d]);
    const float o = (1.f - z) * c + z * h[i];
    h[i] = o;
    if (emb) emb[i] = o;
}

__global__ void fill_zero(float* __restrict__ p, long n)
{
    long i = (long)blockIdx.x * blockDim.x + threadIdx.x;
    const long stride = (long)gridDim.x * blockDim.x;
    for (; i < n; i += stride) p[i] = 0.f;
}

// ---------------------------------------------------------------------------
extern "C" void kernel_launch(void* const* d_in, const int* in_sizes, int n_in,
                              void* d_out, int out_size, void* d_ws, size_t ws_size,
                              hipStream_t stream)
{
    (void)in_sizes; (void)n_in; (void)out_size; (void)ws_size;

    const float* x   = (const float*)d_in[0];
    const int*   src = (const int*)d_in[1];
    const int*   dst = (const int*)d_in[2];
    const int*   ety = (const int*)d_in[3];

    const long NE = (long)NODES * DIM;
    const long Ef = (long)EDGES + NODES;

    // ---- workspace layout (acc, mEnc, den contiguous for one zero-fill) ----
    float* ws   = (float*)d_ws;
    float* hbuf = ws;                 ws += NE;           // persistent h
    float* buf1 = ws;                 ws += NE;           // el / trans[0]
    float* buf2 = ws;                 ws += NE;           // er / trans[1]
    float* gi   = ws;                 ws += (long)NODES * 3 * DIM;
    float* gh   = ws;                 ws += (long)NODES * 3 * DIM;
    float* acc  = ws;                 ws += NE;           // segment-sum target
    unsigned* mEnc = (unsigned*)ws;   ws += NODES;        // encoded segment max
    float* den  = ws;                 ws += NODES;        // softmax denominator
    float* score = ws;                                    // per-edge score / ex

    float* out = (float*)d_out;

    const int edgeWaveBlocks = (int)((Ef + 7) / 8);       // wave-per-edge kernels
    const int edgeThrBlocks  = (int)((Ef + 255) / 256);
    const int elemBlocks     = (int)((NE + 255) / 256);
    const int gemmBlocks64   = ((NODES / 16) * (DIM / 16)       + 7) / 8;
    const int gemmBlocks192  = ((NODES / 16) * (3 * DIM / 16)   + 7) / 8;

    const float* hin = x;
    for (int l = 0; l < 2; ++l) {
        const int base = 4 + 11 * l;
        const float* Wsrc = (const float*)d_in[base + 0];
        const float* bsrc = (const float*)d_in[base + 1];
        const float* Wdst = (const float*)d_in[base + 2];
        const float* bdst = (const float*)d_in[base + 3];
        const float* attn = (const float*)d_in[base + 4];
        const float* ggW  = (const float*)d_in[base + 5];  // (T, D, D)
        const float* ggb  = (const float*)d_in[base + 6];  // (T, D)
        const float* Wih  = (const float*)d_in[base + 7];
        const float* bih  = (const float*)d_in[base + 8];
        const float* Whh  = (const float*)d_in[base + 9];
        const float* bhh  = (const float*)d_in[base + 10];

        // ---------------- GATv2 ----------------
        gemm_bias_wmma<<<gemmBlocks64, 256, 0, stream>>>(hin, Wsrc, bsrc, buf1,
                                                         NODES, DIM, DIM);
        gemm_bias_wmma<<<gemmBlocks64, 256, 0, stream>>>(hin, Wdst, bdst, buf2,
                                                         NODES, DIM, DIM);
        fill_zero<<<2048, 256, 0, stream>>>(acc, NE + 2L * NODES); // acc+mEnc+den
        edge_score<<<edgeWaveBlocks, 256, 0, stream>>>(buf1, buf2, attn, src, dst,
                                                       score, mEnc);
        edge_exp<<<edgeThrBlocks, 256, 0, stream>>>(score, mEnc, den, dst);
        edge_aggregate<<<edgeWaveBlocks, 256, 0, stream>>>(buf1, score, den, src,
                                                           dst, acc);
        elu_write<<<elemBlocks, 256, 0, stream>>>(acc, hbuf, out + (2L * l) * NE,
                                                  NE);

        // ---------------- GatedGraphConv (2 GRU steps) ----------------
        for (int st = 0; st < 2; ++st) {
            gemm_bias_wmma<<<gemmBlocks64, 256, 0, stream>>>(
                hbuf, ggW,               ggb,       buf1, NODES, DIM, DIM);
            gemm_bias_wmma<<<gemmBlocks64, 256, 0, stream>>>(
                hbuf, ggW + DIM * DIM,   ggb + DIM, buf2, NODES, DIM, DIM);
            fill_zero<<<2048, 256, 0, stream>>>(acc, NE);
            ggc_scatter<<<edgeWaveBlocks, 256, 0, stream>>>(buf1, buf2, ety, src,
                                                            dst, acc);
            gemm_bias_wmma<<<gemmBlocks192, 256, 0, stream>>>(
                acc,  Wih, bih, gi, NODES, DIM, 3 * DIM);
            gemm_bias_wmma<<<gemmBlocks192, 256, 0, stream>>>(
                hbuf, Whh, bhh, gh, NODES, DIM, 3 * DIM);
            gru_update<<<elemBlocks, 256, 0, stream>>>(
                gi, gh, hbuf, (st == 1) ? out + (2L * l + 1) * NE : nullptr, NE);
        }
        hin = hbuf;
    }
}